// EmbeddingProposal_47124381171723
// MI455X (gfx1250) — compile-verified
//
#include <hip/hip_runtime.h>
#include <hip/hip_bf16.h>
#include <math.h>

// Problem constants (fixed by the reference)
#define KPART 64     // particles
#define TT    128    // subtrees
#define DD    64     // embedding dim
#define HH    256    // MLP hidden

#define NEG_BIG (-3.0e38f)

typedef float v2f __attribute__((ext_vector_type(2)));
typedef float v8f __attribute__((ext_vector_type(8)));

// ---------------------------------------------------------------------------
// Kernel 1: per-particle Gram matrix via V_WMMA_F32_16X16X4_F32, fused
// distance -> merge log-weights -> gumbel-argmax + online logsumexp epilogue.
// One workgroup (256 threads = 8 waves) per particle k.
// Epilogue is branchless (no EXEC-divergent regions between WMMAs) and uses
// hardware transcendentals (v_log_f32 / v_exp_f32) -- the element count is
// 1M, so libm polynomial log/exp would dominate the whole pipeline.
// ---------------------------------------------------------------------------
__global__ __launch_bounds__(256) void pairdist_sample_kernel(
    const float* __restrict__ Eg,   // [K,T,D]
    const float* __restrict__ Ug,   // [K,T,T] gumbel uniforms
    int*   __restrict__ ws_idx1,
    int*   __restrict__ ws_idx2,
    float* __restrict__ ws_lmp,
    float* __restrict__ out)        // d_out
{
  __shared__ float Elds[TT][DD + 1];   // +1 pad: bank-conflict-free column reads
  __shared__ float norms[TT];
  __shared__ float rCand[256];
  __shared__ int   rIdx[256];
  __shared__ float rMlw[256];
  __shared__ float rMax[256];
  __shared__ float rSum[256];

  const int k    = blockIdx.x;
  const int tid  = threadIdx.x;
  const int lane = tid & 31;
  const int wave = tid >> 5;
  const int half = lane >> 4;   // 0: lanes 0-15, 1: lanes 16-31
  const int lr   = lane & 15;

  // Stage E[k] into LDS (coalesced)
  const float* Ek = Eg + (size_t)k * TT * DD;
  for (int e = tid; e < TT * DD; e += 256)
    Elds[e >> 6][e & 63] = Ek[e];
  __syncthreads();

  // Row squared norms
  if (tid < TT) {
    float s = 0.f;
    #pragma unroll 8
    for (int c = 0; c < DD; ++c) { float v = Elds[tid][c]; s += v * v; }
    norms[tid] = s;
  }
  __syncthreads();

  float bestCand = NEG_BIG; int bestIdx = 0; float bestMlw = NEG_BIG;
  float lseMax = NEG_BIG, lseSum = 0.f;   // finite init: -inf - lseMax stays -inf
  const float* Uk = Ug + (size_t)k * TT * TT;

  const int mi   = wave;               // wave w owns tile-row w of the 8x8 tile grid
  const int mBase = mi * 16 + half * 8;

  // Hoist norms for this wave's 8 output rows (fixed across nj)
  float normM[8];
  #pragma unroll
  for (int r = 0; r < 8; ++r) normM[r] = norms[mBase + r];

  // Prefetch the first tile's gumbel rows (global_prefetch_b8)
  #pragma unroll
  for (int r = 0; r < 8; ++r)
    __builtin_prefetch(Uk + (mBase + r) * TT + lr, 0, 1);

  for (int nj = 0; nj < 8; ++nj) {
    // Prefetch next tile's gumbel rows while the K-loop runs
    const int njn = (nj + 1) & 7;
    #pragma unroll
    for (int r = 0; r < 8; ++r)
      __builtin_prefetch(Uk + (mBase + r) * TT + njn * 16 + lr, 0, 1);

    v8f acc = {0.f,0.f,0.f,0.f,0.f,0.f,0.f,0.f};
    for (int ks = 0; ks < DD / 4; ++ks) {
      const int k0 = ks * 4 + 2 * half;            // A/B K-interleave per ISA layout
      v2f a, b;
      a.x = Elds[mi * 16 + lr][k0];
      a.y = Elds[mi * 16 + lr][k0 + 1];
      b.x = Elds[nj * 16 + lr][k0];                // B = E^T -> same striping from E rows
      b.y = Elds[nj * 16 + lr][k0 + 1];
      acc = __builtin_amdgcn_wmma_f32_16x16x4_f32(
          /*neg_a=*/false, a, /*neg_b=*/false, b,
          /*c_mod=*/(short)0, acc, /*reuse_a=*/false, /*reuse_b=*/false);
    }

    // Fused, branchless epilogue.
    // D tile element (m,n): VGPR r -> M = r + 8*half, lane -> N.
    const int   n  = nj * 16 + lr;
    const float nn = norms[n];
    #pragma unroll
    for (int r = 0; r < 8; ++r) {
      const int m = mBase + r;
      const float g  = acc[r];
      const float sq = normM[r] + nn - 2.f * g;
      const float d  = __builtin_sqrtf(__builtin_fmaxf(sq, 0.f));
      const float mlw = (m == n) ? -INFINITY : -d;           // select, not branch
      const float u   = Uk[m * TT + n];
      const float gum = -__logf(-__logf(u));                 // v_log_f32 x2
      const float cand = mlw + gum;                          // diag: -inf, never max
      const bool better = cand > bestCand;
      bestIdx  = better ? (m * TT + n) : bestIdx;            // v_cndmask
      bestMlw  = better ? mlw : bestMlw;
      bestCand = better ? cand : bestCand;
      // Branchless online logsumexp; diagonal contributes exp(-inf)=0.
      const float newMax = __builtin_fmaxf(lseMax, mlw);
      lseSum = lseSum * __expf(lseMax - newMax) + __expf(mlw - newMax);
      lseMax = newMax;
    }
  }

  rCand[tid] = bestCand; rIdx[tid] = bestIdx; rMlw[tid] = bestMlw;
  rMax[tid]  = lseMax;   rSum[tid] = lseSum;
  __syncthreads();
  for (int s = 128; s > 0; s >>= 1) {
    if (tid < s) {
      if (rCand[tid + s] > rCand[tid]) {
        rCand[tid] = rCand[tid + s]; rIdx[tid] = rIdx[tid + s]; rMlw[tid] = rMlw[tid + s];
      }
      const float m1 = rMax[tid], m2 = rMax[tid + s];
      const float M  = __builtin_fmaxf(m1, m2);
      rSum[tid] = rSum[tid] * __expf(m1 - M) + rSum[tid + s] * __expf(m2 - M);
      rMax[tid] = M;
    }
    __syncthreads();
  }
  if (tid == 0) {
    const float lse = rMax[0] + logf(rSum[0]);               // precise tail (64 total)
    const float lmp = rMlw[0] + 0.6931471805599453f - lse;   // + log(2)
    const int bi = rIdx[0];
    const int i1 = bi / TT, i2 = bi % TT;
    ws_idx1[k] = i1; ws_idx2[k] = i2; ws_lmp[k] = lmp;
    out[k]         = (float)i1;
    out[KPART + k] = (float)i2;
  }
}

// ---------------------------------------------------------------------------
// Kernel 2: gather X = concat(child1, child2) [64x128], GEMM1 via WMMA:
// H = relu(X @ W1 + b1) [64x256] -> workspace. Single 256-thread block.
// ---------------------------------------------------------------------------
__global__ __launch_bounds__(256) void mlp_gemm1_kernel(
    const float* __restrict__ Eg,
    const float* __restrict__ W1,   // [128,256] row-major
    const float* __restrict__ b1,   // [256]
    const int*   __restrict__ ws_idx1,
    const int*   __restrict__ ws_idx2,
    float* __restrict__ Xws,        // [64,128]
    float* __restrict__ Hws)        // [64,256]
{
  const int XS = 129;               // padded stride -> conflict-free A-frag loads
  __shared__ float Xs[KPART * 129];

  const int tid  = threadIdx.x;
  const int lane = tid & 31, wave = tid >> 5;
  const int half = lane >> 4, lr = lane & 15;

  // Gather children rows into X (and mirror to global for the tail kernel)
  for (int e = tid; e < KPART * 2 * DD; e += 256) {
    const int row = e >> 7, col = e & 127;
    const int src = (col < DD) ? ws_idx1[row] : ws_idx2[row];
    const int c   = (col < DD) ? col : col - DD;
    const float v = Eg[(size_t)row * TT * DD + (size_t)src * DD + c];
    Xs[row * XS + col] = v;
    Xws[row * (2 * DD) + col] = v;
  }
  __syncthreads();

  // 4 x 16 grid of 16x16 tiles, 8 waves x 8 tiles, 32 k-steps of K=4
  for (int i = 0; i < 8; ++i) {
    const int tt = wave * 8 + i;
    const int mi = tt >> 4, nj = tt & 15;
    v8f acc = {0.f,0.f,0.f,0.f,0.f,0.f,0.f,0.f};
    for (int ks = 0; ks < (2 * DD) / 4; ++ks) {
      const int k0 = ks * 4 + 2 * half;
      v2f a, b;
      a.x = Xs[(mi * 16 + lr) * XS + k0];
      a.y = Xs[(mi * 16 + lr) * XS + k0 + 1];
      b.x = W1[k0 * HH + nj * 16 + lr];
      b.y = W1[(k0 + 1) * HH + nj * 16 + lr];
      acc = __builtin_amdgcn_wmma_f32_16x16x4_f32(false, a, false, b, (short)0, acc, false, false);
    }
    #pragma unroll
    for (int r = 0; r < 8; ++r) {
      const int m = mi * 16 + half * 8 + r;
      const int n = nj * 16 + lr;
      Hws[m * HH + n] = __builtin_fmaxf(acc[r] + b1[n], 0.f);   // bias + relu
    }
  }
}

// ---------------------------------------------------------------------------
// Kernel 3: GEMM2 via WMMA: emb = H @ W2 + b2 [64x64] -> d_out, then the
// per-particle branch-length sampling tail. Single 256-thread block.
// ---------------------------------------------------------------------------
__global__ __launch_bounds__(256) void mlp_gemm2_tail_kernel(
    const float* __restrict__ Hws,  // [64,256]
    const float* __restrict__ W2,   // [256,64]
    const float* __restrict__ b2,   // [64]
    const float* __restrict__ Xws,  // [64,128] gathered children
    const float* __restrict__ u1,
    const float* __restrict__ u2,
    const float* __restrict__ ws_lmp,
    float* __restrict__ out)
{
  const int ES = 65;
  __shared__ float Es[KPART * 65];  // merged embeddings, padded

  const int tid  = threadIdx.x;
  const int lane = tid & 31, wave = tid >> 5;
  const int half = lane >> 4, lr = lane & 15;

  // 4 x 4 grid of 16x16 tiles, 8 waves x 2 tiles, 64 k-steps of K=4
  for (int i = 0; i < 2; ++i) {
    const int tt = wave * 2 + i;
    const int mi = tt >> 2, nj = tt & 3;
    v8f acc = {0.f,0.f,0.f,0.f,0.f,0.f,0.f,0.f};
    for (int ks = 0; ks < HH / 4; ++ks) {
      const int k0 = ks * 4 + 2 * half;
      v2f a, b;
      a.x = Hws[(mi * 16 + lr) * HH + k0];       // L2-hot, 64 KB total
      a.y = Hws[(mi * 16 + lr) * HH + k0 + 1];
      b.x = W2[k0 * DD + nj * 16 + lr];
      b.y = W2[(k0 + 1) * DD + nj * 16 + lr];
      acc = __builtin_amdgcn_wmma_f32_16x16x4_f32(false, a, false, b, (short)0, acc, false, false);
    }
    #pragma unroll
    for (int r = 0; r < 8; ++r) {
      const int m = mi * 16 + half * 8 + r;
      const int n = nj * 16 + lr;
      const float v = acc[r] + b2[n];
      Es[m * ES + n] = v;
      out[256 + m * DD + n] = v;                 // embedding_KxD
    }
  }
  __syncthreads();

  // Branch sampling tail: one thread per particle (precise libm, 64 lanes total)
  if (tid < KPART) {
    const int k = tid;
    float d1 = 0.f, d2 = 0.f;
    #pragma unroll 8
    for (int c = 0; c < DD; ++c) {
      const float e = Es[k * ES + c];
      const float a = Xws[k * (2 * DD) + c]      - e;
      const float b = Xws[k * (2 * DD) + DD + c] - e;
      d1 += a * a; d2 += b * b;
    }
    const float dist1 = sqrtf(d1), dist2 = sqrtf(d2);
    const float rate1 = 1.f / (dist1 + 1e-4f);
    const float rate2 = 1.f / (dist2 + 1e-4f);
    const float br1 = -logf(u1[k]) / rate1;      // exponential sample
    const float br2 = -logf(u2[k]) / rate2;
    const float lb1 = logf(rate1) - rate1 * br1;
    const float lb2 = logf(rate2) - rate2 * br2;
    out[128 + k]  = br1;
    out[192 + k]  = br2;
    out[4352 + k] = ws_lmp[k] + lb1 + lb2;       // log_v_plus
  }
}

// ---------------------------------------------------------------------------
extern "C" void kernel_launch(void* const* d_in, const int* in_sizes, int n_in,
                              void* d_out, int out_size, void* d_ws, size_t ws_size,
                              hipStream_t stream) {
  // setup_inputs order:
  // 0:N 1:leaf_counts 2:embeddings 3:log_felsensteins 4:site_positions
  // 5:W1 6:b1 7:W2 8:b2 9:gumbel_u 10:uniform1 11:uniform2
  const float* Eg = (const float*)d_in[2];
  const float* W1 = (const float*)d_in[5];
  const float* b1 = (const float*)d_in[6];
  const float* W2 = (const float*)d_in[7];
  const float* b2 = (const float*)d_in[8];
  const float* Ug = (const float*)d_in[9];
  const float* u1 = (const float*)d_in[10];
  const float* u2 = (const float*)d_in[11];
  float* out = (float*)d_out;

  // Workspace layout
  int*   ws_idx1 = (int*)d_ws;                       // [64]
  int*   ws_idx2 = ws_idx1 + KPART;                  // [64]
  float* ws_lmp  = (float*)(ws_idx2 + KPART);        // [64]
  float* Xws     = ws_lmp + KPART;                   // [64*128]
  float* Hws     = Xws + KPART * 2 * DD;             // [64*256]

  pairdist_sample_kernel<<<KPART, 256, 0, stream>>>(Eg, Ug, ws_idx1, ws_idx2, ws_lmp, out);
  mlp_gemm1_kernel<<<1, 256, 0, stream>>>(Eg, W1, b1, ws_idx1, ws_idx2, Xws, Hws);
  mlp_gemm2_tail_kernel<<<1, 256, 0, stream>>>(Hws, W2, b2, Xws, u1, u2, ws_lmp, out);
}